// GCNBiasNodeClassifier_26731876451146
// MI455X (gfx1250) — compile-verified
//
#include <hip/hip_runtime.h>

#define NNODES 50000
#define NEDGES 600000
#define HDIM   128
#define MTILES (NNODES / 16)   // 3125, exact
#define BSTR   130             // LDS row stride (even -> 8B-aligned b64 reads, bank spread)

typedef float v2f __attribute__((ext_vector_type(2)));
typedef float v8f __attribute__((ext_vector_type(8)));

// ---------------------------------------------------------------------------
// C[N x 128] = A[N x 128] * B[128 x 128]   (+ optional bias & ReLU epilogue)
// One wave computes a 16-row x 128-col strip via V_WMMA_F32_16X16X4_F32.
// B is staged TRANSPOSED in LDS (Bs[n][k], stride 130) so each lane's
// (K, K+1) fragment is one aligned ds_load_b64 into an even VGPR pair.
// ---------------------------------------------------------------------------
__global__ __launch_bounds__(128) void gemm128_wmma(
    const float* __restrict__ A, const float* __restrict__ B,
    const float* __restrict__ bias, float* __restrict__ C,
    int nTiles, int fuseBiasRelu)
{
    __shared__ float Bs[HDIM * BSTR];          // ~65 KB, Bs[n*BSTR + k] = B[k][n]
    {
        const float4* s4 = (const float4*)B;
        for (int i = threadIdx.x; i < HDIM * HDIM / 4; i += 128) {
            const int r = i >> 5;              // K row
            const int c = (i & 31) * 4;        // N col base
            float4 v = s4[i];
            Bs[(c + 0) * BSTR + r] = v.x;
            Bs[(c + 1) * BSTR + r] = v.y;
            Bs[(c + 2) * BSTR + r] = v.z;
            Bs[(c + 3) * BSTR + r] = v.w;
        }
    }
    __syncthreads();

    const int wave  = threadIdx.x >> 5;
    const int lane  = threadIdx.x & 31;
    const int tileM = blockIdx.x * 4 + wave;
    if (tileM >= nTiles) return;               // wave-uniform: EXEC stays full

    const int mn   = lane & 15;                // A row within tile / B,C column
    const int half = lane >> 4;                // 0 or 1
    const int kb   = half * 2;                 // K sub-offset within group of 4

    const float* arow = A + (size_t)(tileM * 16 + mn) * HDIM;

    v8f acc[8] = {};
    for (int k = 0; k < HDIM; k += 4) {
        // A fragment 16x4 f32: lanes 0-15 hold K=k..k+1, lanes 16-31 K=k+2..k+3
        v2f a = *(const v2f*)(arow + k + kb);          // global b64, 8B aligned
        const float* bcol = Bs + k + kb;
#pragma unroll
        for (int t = 0; t < 8; ++t) {
            const int n = t * 16 + mn;
            v2f b = *(const v2f*)(bcol + n * BSTR);    // single ds_load_b64
            acc[t] = __builtin_amdgcn_wmma_f32_16x16x4_f32(
                false, a, false, b, (short)0, acc[t], false, false);
        }
    }

    // C/D 16x16 f32 layout: VGPR i -> row (half*8 + i), col = mn
    const int row0 = tileM * 16 + half * 8;
#pragma unroll
    for (int t = 0; t < 8; ++t) {
        const int n  = t * 16 + mn;
        const float bv = fuseBiasRelu ? bias[n] : 0.0f;
#pragma unroll
        for (int i = 0; i < 8; ++i) {
            float v = acc[t][i];
            if (fuseBiasRelu) v = fmaxf(v + bv, 0.0f);
            C[(size_t)(row0 + i) * HDIM + n] = v;
        }
    }
}

// ---------------------------------------------------------------------------
// Degree / normalization helpers
// ---------------------------------------------------------------------------
__global__ void init_deg(float* deg, int n) {
    int i = blockIdx.x * blockDim.x + threadIdx.x;
    if (i < n) deg[i] = 1.0f;                      // self-loop
}

__global__ void count_deg(const long long* __restrict__ dst, float* deg, int e) {
    int i = blockIdx.x * blockDim.x + threadIdx.x;
    if (i < e) atomicAdd(&deg[(int)dst[i]], 1.0f);
}

__global__ void to_rsqrt(float* deg, int n) {
    int i = blockIdx.x * blockDim.x + threadIdx.x;
    if (i < n) deg[i] = rsqrtf(deg[i]);
}

// agg[n][j] = bias[j]  (scatter then accumulates messages on top)
__global__ void init_agg(float* __restrict__ agg, const float* __restrict__ b, int total) {
    int i = blockIdx.x * blockDim.x + threadIdx.x;
    if (i < total) agg[i] = b[i & (HDIM - 1)];
}

// ---------------------------------------------------------------------------
// Edge scatter: one wave32 per (edge | self-loop); float4 per lane covers 128.
// ---------------------------------------------------------------------------
__global__ __launch_bounds__(256) void scatter_edges(
    const float* __restrict__ hW, const long long* __restrict__ src,
    const long long* __restrict__ dst, const float* __restrict__ dinv,
    float* __restrict__ agg, int ne, int nn)
{
    const int e    = (blockIdx.x * blockDim.x + threadIdx.x) >> 5;
    const int lane = threadIdx.x & 31;
    if (e >= ne + nn) return;

    int s, d; float w;
    if (e < ne) {
        s = (int)src[e]; d = (int)dst[e];
        w = dinv[s] * dinv[d];
    } else {
        s = d = e - ne;
        float di = dinv[s];
        w = di * di;
    }
    float4 v = *((const float4*)(hW + (size_t)s * HDIM) + lane);
    float* out = agg + (size_t)d * HDIM + lane * 4;
    atomicAdd(out + 0, v.x * w);
    atomicAdd(out + 1, v.y * w);
    atomicAdd(out + 2, v.z * w);
    atomicAdd(out + 3, v.w * w);
}

// ---------------------------------------------------------------------------
// Fused LayerNorm + ReLU: one wave32 per node, shuffle reduction.
// ---------------------------------------------------------------------------
__global__ __launch_bounds__(256) void ln_relu(
    const float* __restrict__ agg, const float* __restrict__ g,
    const float* __restrict__ be, float* __restrict__ out, int n)
{
    const int node = (blockIdx.x * blockDim.x + threadIdx.x) >> 5;
    const int lane = threadIdx.x & 31;
    if (node >= n) return;

    float4 v = *((const float4*)(agg + (size_t)node * HDIM) + lane);
    float s  = v.x + v.y + v.z + v.w;
    float sq = v.x * v.x + v.y * v.y + v.z * v.z + v.w * v.w;
#pragma unroll
    for (int off = 16; off; off >>= 1) {
        s  += __shfl_xor(s, off, 32);
        sq += __shfl_xor(sq, off, 32);
    }
    const float mu  = s * (1.0f / HDIM);
    const float var = sq * (1.0f / HDIM) - mu * mu;
    const float rs  = rsqrtf(var + 1e-5f);

    float4 gv = ((const float4*)g)[lane];
    float4 bv = ((const float4*)be)[lane];
    float4 o;
    o.x = fmaxf((v.x - mu) * rs * gv.x + bv.x, 0.0f);
    o.y = fmaxf((v.y - mu) * rs * gv.y + bv.y, 0.0f);
    o.z = fmaxf((v.z - mu) * rs * gv.z + bv.z, 0.0f);
    o.w = fmaxf((v.w - mu) * rs * gv.w + bv.w, 0.0f);
    *((float4*)(out + (size_t)node * HDIM) + lane) = o;
}

// ---------------------------------------------------------------------------
// Head output: logits[n] = dot(h[n], hW2) + hb2  (C == 1)
// ---------------------------------------------------------------------------
__global__ __launch_bounds__(256) void head_out(
    const float* __restrict__ h, const float* __restrict__ w2,
    const float* __restrict__ b2, float* __restrict__ out, int n)
{
    const int node = (blockIdx.x * blockDim.x + threadIdx.x) >> 5;
    const int lane = threadIdx.x & 31;
    if (node >= n) return;

    float4 v = *((const float4*)(h + (size_t)node * HDIM) + lane);
    float4 w = ((const float4*)w2)[lane];
    float s = v.x * w.x + v.y * w.y + v.z * w.z + v.w * w.w;
#pragma unroll
    for (int off = 16; off; off >>= 1) s += __shfl_xor(s, off, 32);
    if (lane == 0) out[node] = s + b2[0];
}

// ---------------------------------------------------------------------------
extern "C" void kernel_launch(void* const* d_in, const int* in_sizes, int n_in,
                              void* d_out, int out_size, void* d_ws, size_t ws_size,
                              hipStream_t stream)
{
    (void)in_sizes; (void)n_in; (void)out_size; (void)ws_size;

    const float*     x   = (const float*)d_in[0];
    const long long* ei  = (const long long*)d_in[1];
    const long long* src = ei;
    const long long* dst = ei + NEDGES;

    const float* W[3]  = { (const float*)d_in[2],  (const float*)d_in[6],  (const float*)d_in[10] };
    const float* b[3]  = { (const float*)d_in[3],  (const float*)d_in[7],  (const float*)d_in[11] };
    const float* g[3]  = { (const float*)d_in[4],  (const float*)d_in[8],  (const float*)d_in[12] };
    const float* be[3] = { (const float*)d_in[5],  (const float*)d_in[9],  (const float*)d_in[13] };
    const float* hW1 = (const float*)d_in[14];
    const float* hb1 = (const float*)d_in[15];
    const float* hW2 = (const float*)d_in[16];
    const float* hb2 = (const float*)d_in[17];

    const size_t NB = (size_t)NNODES * HDIM;
    float* buf0 = (float*)d_ws;       // hW  scratch
    float* buf1 = buf0 + NB;          // agg scratch
    float* buf2 = buf1 + NB;          // activation scratch
    float* dinv = buf2 + NB;          // N floats

    // --- GCN symmetric normalization ---
    init_deg <<<(NNODES + 255) / 256, 256, 0, stream>>>(dinv, NNODES);
    count_deg<<<(NEDGES + 255) / 256, 256, 0, stream>>>(dst, dinv, NEDGES);
    to_rsqrt <<<(NNODES + 255) / 256, 256, 0, stream>>>(dinv, NNODES);

    const int gemmGrid    = (MTILES + 3) / 4;                    // 4 waves / block
    const int aggElems    = (int)NB;
    const int scatterGrid = (NEDGES + NNODES + 7) / 8;           // 8 waves / block
    const int waveGrid    = (NNODES + 7) / 8;

    const float* hin = x;
    for (int l = 0; l < 3; ++l) {
        gemm128_wmma <<<gemmGrid, 128, 0, stream>>>(hin, W[l], b[l], buf0, MTILES, 0);
        init_agg     <<<(aggElems + 255) / 256, 256, 0, stream>>>(buf1, b[l], aggElems);
        scatter_edges<<<scatterGrid, 256, 0, stream>>>(buf0, src, dst, dinv, buf1,
                                                       NEDGES, NNODES);
        ln_relu      <<<waveGrid, 256, 0, stream>>>(buf1, g[l], be[l], buf2, NNODES);
        hin = buf2;
    }

    // --- MLP head ---
    gemm128_wmma<<<gemmGrid, 128, 0, stream>>>(buf2, hW1, hb1, buf0, MTILES, 1);
    head_out    <<<waveGrid, 256, 0, stream>>>(buf0, hW2, hb2, (float*)d_out, NNODES);
}